// FunctionalConnectivityGraphBuilder_80539226735382
// MI455X (gfx1250) — compile-verified
//
#include <hip/hip_runtime.h>
#include <hip/hip_bf16.h>

typedef __attribute__((ext_vector_type(2))) float v2f;
typedef __attribute__((ext_vector_type(8))) float v8f;
typedef __attribute__((ext_vector_type(4))) unsigned int u32x4;
typedef __attribute__((ext_vector_type(8))) int i32x8;
typedef __attribute__((ext_vector_type(4))) int i32x4;

#define RN 96
#define TN 2048
#define NBQ 5
#define NBANDS 160           // B*NB = 32*5
#define NROWS (NBANDS * RN)  // 15360
#define N_TRIL 4560          // 96*95/2
#define SORT_N 8192
#define FLT_MAX_C 3.402823466e38f
#define PI_F 3.14159265358979f

#define KSLAB 32
#define LDS_ROW 33          // 32 + 1 pad DWORD (matches TDM pad_interval=32,pad_amount=1)
#define SLAB_BYTES (RN * LDS_ROW * 4)

// ---------------------------------------------------------------------------
// Kernel 1: per-row statistics (mean, std ddof=1, energy, maxamp, norm)
// stats layout: stats[rid*8 + {0:mean,1:std,2:energy,3:maxamp,4:ent,5:norm}]
// ---------------------------------------------------------------------------
__global__ __launch_bounds__(256)
void stats_kernel(const float* __restrict__ wc, float* __restrict__ stats) {
    const int rid  = blockIdx.x;          // band*96 + r
    const int band = rid / RN;
    const int r    = rid - band * RN;
    const int bi   = band / NBQ;
    const int nb   = band - bi * NBQ;
    const float* x = wc + (((size_t)bi * RN + r) * NBQ + nb) * TN;

    __shared__ float s_sum[256], s_sq[256], s_mx[256];
    const int tid = threadIdx.x;
    float sum = 0.f, sq = 0.f, mx = 0.f;
    for (int i = tid; i < TN; i += 256) {
        float v = x[i];
        sum += v; sq += v * v; mx = fmaxf(mx, fabsf(v));
    }
    s_sum[tid] = sum; s_sq[tid] = sq; s_mx[tid] = mx;
    __syncthreads();
    for (int s = 128; s > 0; s >>= 1) {
        if (tid < s) {
            s_sum[tid] += s_sum[tid + s];
            s_sq[tid]  += s_sq[tid + s];
            s_mx[tid]   = fmaxf(s_mx[tid], s_mx[tid + s]);
        }
        __syncthreads();
    }
    if (tid == 0) {
        float total = s_sum[0], sumsq = s_sq[0];
        float mean   = total / (float)TN;
        float varnum = fmaxf(sumsq - (float)TN * mean * mean, 0.f);
        float sd     = sqrtf(varnum / (float)(TN - 1));
        float nrm    = sqrtf(varnum);
        if (nrm == 0.f) nrm = 1.f;
        float* o = stats + (size_t)rid * 8;
        o[0] = mean; o[1] = sd; o[2] = sumsq; o[3] = s_mx[0]; o[4] = 0.f; o[5] = nrm;
    }
}

// ---------------------------------------------------------------------------
// Kernel 2: 2048-pt radix-2 FFT in LDS -> spectral entropy of first 1024 bins
// ---------------------------------------------------------------------------
__global__ __launch_bounds__(256)
void fft_entropy_kernel(const float* __restrict__ wc, float* __restrict__ stats) {
    const int rid  = blockIdx.x;
    const int band = rid / RN;
    const int r    = rid - band * RN;
    const int bi   = band / NBQ;
    const int nb   = band - bi * NBQ;
    const float* x = wc + (((size_t)bi * RN + r) * NBQ + nb) * TN;

    __shared__ float re[TN], im[TN];
    __shared__ float red[256];
    const int tid = threadIdx.x;

    // load in bit-reversed order (11-bit reversal)
    for (int i = tid; i < TN; i += 256) {
        unsigned j = __brev((unsigned)i) >> 21;
        re[j] = x[i]; im[j] = 0.f;
    }
    __syncthreads();

    for (int half = 1; half < TN; half <<= 1) {
        for (int b = tid; b < TN / 2; b += 256) {
            int pos = b & (half - 1);
            int i0  = ((b - pos) << 1) + pos;
            int i1  = i0 + half;
            float ang = -PI_F * (float)pos / (float)half;
            float wr = __cosf(ang), wi = __sinf(ang);
            float xr = re[i1], xi = im[i1];
            float tr = wr * xr - wi * xi;
            float ti = wr * xi + wi * xr;
            float ur = re[i0], ui = im[i0];
            re[i1] = ur - tr; im[i1] = ui - ti;
            re[i0] = ur + tr; im[i0] = ui + ti;
        }
        __syncthreads();
    }

    float ls = 0.f;
    for (int i = tid; i < TN / 2; i += 256) ls += re[i] * re[i] + im[i] * im[i];
    red[tid] = ls; __syncthreads();
    for (int s = 128; s > 0; s >>= 1) { if (tid < s) red[tid] += red[tid + s]; __syncthreads(); }
    float ssum = red[0];
    __syncthreads();
    if (ssum == 0.f) ssum = 1.f;
    float inv = 1.f / ssum;
    float le = 0.f;
    for (int i = tid; i < TN / 2; i += 256) {
        float p = (re[i] * re[i] + im[i] * im[i]) * inv;
        le += p * __logf(p + 1e-10f);
    }
    red[tid] = le; __syncthreads();
    for (int s = 128; s > 0; s >>= 1) { if (tid < s) red[tid] += red[tid + s]; __syncthreads(); }
    if (tid == 0) stats[(size_t)rid * 8 + 4] = -red[0];
}

// ---------------------------------------------------------------------------
// Kernel 3: raw Gram G = X @ X^T per band via V_WMMA_F32_16X16X4_F32,
// with K-slabs DMA'd into double-buffered LDS by the Tensor Data Mover.
// blockDim = 192 (6 waves); wave w owns output tile-row w (6 tiles of 16x16).
// ---------------------------------------------------------------------------
__global__ __launch_bounds__(192)
void gram_kernel(const float* __restrict__ wc, float* __restrict__ G) {
    const int band = blockIdx.x;          // 0..159
    const int bi   = band / NBQ;
    const int nb   = band - bi * NBQ;
    const float* base = wc + (size_t)(bi * (RN * NBQ) + nb) * TN; // row r -> base + r*NBQ*TN

    __shared__ float xs[2][RN][LDS_ROW];  // double-buffered 32-wide K slabs

    const int tid  = threadIdx.x;
    const int wave = tid >> 5;            // 0..5 -> tile row
    const int lane = tid & 31;
    const int mr   = lane & 15;           // M (and N) within tile
    const int kh   = (lane >> 4) << 1;    // K offset: 0 (lanes 0-15) or 2 (lanes 16-31)

    const unsigned long long gbase = (unsigned long long)(uintptr_t)base;
    // LDS aperture keeps the byte offset in addr[31:0] (ISA 10.2)
    const unsigned lds0 = (unsigned)(uintptr_t)&xs[0][0][0];

    // Issue a TDM load of a 96x32 f32 tile (row stride 10240 elems) into LDS,
    // with hardware padding of 1 DWORD per 32 DWORDs (-> LDS row stride 33).
    auto tdm_issue = [&](int buf, int kb) {
        unsigned long long ga = gbase + (unsigned long long)kb * 4ull;
        unsigned lds = lds0 + (unsigned)buf * (unsigned)SLAB_BYTES;
        u32x4 g0;
        g0[0] = 1u;                                           // count=1, user desc
        g0[1] = lds;                                          // lds_addr
        g0[2] = (unsigned)(ga & 0xFFFFFFFFull);               // global_addr[31:0]
        g0[3] = (unsigned)((ga >> 32) & 0x01FFFFFFull)        // global_addr[56:32]
              | (2u << 30);                                   // type=2 (image)
        i32x8 g1;
        g1[0] = (int)((2u << 16)      // data_size = 4 bytes
                    | (1u << 20)      // pad_enable
                    | (4u << 22)      // pad_interval: 32 DWORDs
                    | (0u << 25));    // pad_amount: 1 DWORD
        g1[1] = (int)((2048u & 0xFFFFu) << 16);               // tensor_dim0[15:0]
        g1[2] = (int)((2048u >> 16) | ((unsigned)RN << 16));  // tensor_dim0 hi | tensor_dim1 lo
        g1[3] = (int)((unsigned)KSLAB << 16);                 // tensor_dim1 hi | tile_dim0
        g1[4] = (int)RN;                                      // tile_dim1 | tile_dim2=0
        g1[5] = (int)(NBQ * TN);                              // tensor_dim0_stride = 10240
        g1[6] = 0;                                            // stride0 hi | stride1 lo
        g1[7] = 0;
        i32x4 gz4 = {0, 0, 0, 0};                             // 2-D tensor: groups 2/3 unused
        i32x8 gz8 = {0, 0, 0, 0, 0, 0, 0, 0};                 // extra group (clang-23 form)
        __builtin_amdgcn_tensor_load_to_lds(g0, g1, gz4, gz4, gz8, 0);
    };

    v8f acc[6];
#pragma unroll
    for (int t = 0; t < 6; ++t)
#pragma unroll
        for (int e = 0; e < 8; ++e) acc[t][e] = 0.f;

    const int rowA = wave * 16 + mr;
    const bool leader = (wave == 0);

    if (leader) tdm_issue(0, 0);

    int cur = 0;
    for (int it = 0; it < TN / KSLAB; ++it) {
        __syncthreads();   // everyone done with buf[1-cur] -> safe for TDM to overwrite
        if (leader) {
            if (it + 1 < TN / KSLAB) {
                tdm_issue(1 - cur, (it + 1) * KSLAB);
                __builtin_amdgcn_s_wait_tensorcnt(1);  // oldest (buf[cur]) complete
            } else {
                __builtin_amdgcn_s_wait_tensorcnt(0);
            }
        }
        __syncthreads();   // buf[cur] ready for all waves
#pragma unroll
        for (int k = 0; k < KSLAB; k += 4) {
            v2f a;
            a.x = xs[cur][rowA][k + kh];
            a.y = xs[cur][rowA][k + kh + 1];
#pragma unroll
            for (int tn = 0; tn < 6; ++tn) {
                v2f bfr;
                const int rowB = tn * 16 + mr;
                bfr.x = xs[cur][rowB][k + kh];
                bfr.y = xs[cur][rowB][k + kh + 1];
                acc[tn] = __builtin_amdgcn_wmma_f32_16x16x4_f32(
                    false, a, false, bfr, (short)0, acc[tn], false, false);
            }
        }
        cur ^= 1;
    }

    // C/D layout: VGPR v, lanes 0-15 -> M=v, N=lane; lanes 16-31 -> M=v+8, N=lane-16
    float* Gb = G + (size_t)band * (RN * RN);
    const int mhi = (lane >> 4) << 3;
#pragma unroll
    for (int tn = 0; tn < 6; ++tn) {
#pragma unroll
        for (int v = 0; v < 8; ++v) {
            int M = wave * 16 + v + mhi;
            int N = tn * 16 + mr;
            Gb[M * RN + N] = acc[tn][v];
        }
    }
}

// ---------------------------------------------------------------------------
// Kernel 4: per-band finalize: C matrix, quantile (bitonic sort), A output,
// graph properties, node features, MLP.
// ---------------------------------------------------------------------------
__global__ __launch_bounds__(256)
void finalize_kernel(const float* __restrict__ G, const float* __restrict__ stats,
                     const int* __restrict__ community,
                     const float* __restrict__ W1, const float* __restrict__ b1,
                     const float* __restrict__ W2, const float* __restrict__ b2,
                     float* __restrict__ outA, float* __restrict__ outF) {
    const int band = blockIdx.x;
    __shared__ float pool[RN * RN];       // sort array (8192) then Bm (9216)
    __shared__ float meanv[RN], normv[RN], degs[RN];
    __shared__ int   comm[RN];
    __shared__ float red[256];
    __shared__ float sc[8];
    __shared__ float feat[11];
    __shared__ float h[32];

    const int tid = threadIdx.x;
    if (tid < RN) {
        const float* s = stats + (size_t)(band * RN + tid) * 8;
        meanv[tid] = s[0];
        normv[tid] = s[5];
        comm[tid]  = community[tid];
    }
    __syncthreads();

    const float* Gb = G + (size_t)band * (RN * RN);

    // correlation value computed on the fly (keeps LDS small)
    auto cval = [&](int i, int j) -> float {
        if (i == j) return 0.f;
        float c = (Gb[i * RN + j] - (float)TN * meanv[i] * meanv[j]) / (normv[i] * normv[j]);
        if (!(c == c)) c = 0.f;                       // nan_to_num(nan=0)
        else if (c >  FLT_MAX_C) c =  FLT_MAX_C;
        else if (c < -FLT_MAX_C) c = -FLT_MAX_C;
        return c;
    };

    // fill sort array with |C| of strict lower triangle, pad with +FLT_MAX
    for (int idx = tid; idx < SORT_N; idx += 256) pool[idx] = FLT_MAX_C;
    __syncthreads();
    for (int idx = tid; idx < RN * RN; idx += 256) {
        int i = idx / RN, j = idx - i * RN;
        if (i > j) pool[(i * (i - 1)) / 2 + j] = fabsf(cval(i, j));
    }
    __syncthreads();

    // bitonic sort ascending over 8192 elements
    for (int k = 2; k <= SORT_N; k <<= 1) {
        for (int j = k >> 1; j > 0; j >>= 1) {
            for (int i = tid; i < SORT_N; i += 256) {
                int ixj = i ^ j;
                if (ixj > i) {
                    float a = pool[i], bv = pool[ixj];
                    bool up = ((i & k) == 0);
                    if (up ? (a > bv) : (a < bv)) { pool[i] = bv; pool[ixj] = a; }
                }
            }
            __syncthreads();
        }
    }

    if (tid == 0) {
        float pos  = 0.8f * (float)(N_TRIL - 1);   // 3647.2
        int   lo   = (int)pos;
        float frac = pos - (float)lo;
        sc[0] = pool[lo] + frac * (pool[lo + 1] - pool[lo]);
    }
    __syncthreads();
    const float thr = sc[0];
    __syncthreads();

    // write A, build Bm (reuse pool)
    float* Bm = pool;
    for (int idx = tid; idx < RN * RN; idx += 256) {
        int i = idx / RN, j = idx - i * RN;
        float c = cval(i, j);
        float a = (fabsf(c) >= thr) ? c : 0.f;
        outA[(size_t)band * (RN * RN) + idx] = a;
        Bm[idx] = (a != 0.f) ? 1.f : 0.f;
    }
    __syncthreads();

    if (tid < RN) {
        float d = 0.f;
        for (int j = 0; j < RN; ++j) d += Bm[tid * RN + j];
        degs[tid] = d;
    }
    __syncthreads();
    if (tid == 0) {
        float sd = 0.f, poss = 0.f;
        for (int i = 0; i < RN; ++i) { sd += degs[i]; poss += degs[i] * (degs[i] - 1.f); }
        sc[1] = 0.5f * sd;     // ne
        sc[2] = 0.5f * poss;   // poss
        sc[3] = sd / (float)RN;// avg_deg
    }
    __syncthreads();
    const float ne      = sc[1];
    const float poss    = sc[2];
    const float avg_deg = sc[3];
    const float m2      = 2.f * ne;
    const float m2s     = (m2 > 0.f) ? m2 : 1.f;

    float ltri = 0.f, leff = 0.f, lmod = 0.f;
    for (int idx = tid; idx < RN * RN; idx += 256) {
        int i = idx / RN, j = idx - i * RN;
        float bij = Bm[idx];
        if (bij != 0.f) {
            float dot = 0.f;
            const float* Ri = Bm + i * RN;
            const float* Rj = Bm + j * RN;   // Bm symmetric: row j == column j
            for (int kk = 0; kk < RN; ++kk) dot += Ri[kk] * Rj[kk];
            ltri += dot;
        }
        float d = bij + ((i == j) ? 1.f : 0.f);
        if (d > 0.f) leff += 1.f / d;
        if ((comm[i] == comm[j]) && (i != j)) lmod += bij - degs[i] * degs[j] / m2s;
    }

    red[tid] = ltri; __syncthreads();
    for (int s = 128; s > 0; s >>= 1) { if (tid < s) red[tid] += red[tid + s]; __syncthreads(); }
    float tri6 = red[0] / 6.f; __syncthreads();
    red[tid] = leff; __syncthreads();
    for (int s = 128; s > 0; s >>= 1) { if (tid < s) red[tid] += red[tid + s]; __syncthreads(); }
    float effsum = red[0]; __syncthreads();
    red[tid] = lmod; __syncthreads();
    for (int s = 128; s > 0; s >>= 1) { if (tid < s) red[tid] += red[tid + s]; __syncthreads(); }
    float modsum = red[0]; __syncthreads();

    if (tid == 0) {
        float n0 = 0, n1 = 0, n2 = 0, n3 = 0, n4 = 0;
        for (int i = 0; i < RN; ++i) {
            const float* s = stats + (size_t)(band * RN + i) * 8;
            n0 += s[0]; n1 += s[1]; n2 += s[2]; n3 += s[3]; n4 += s[4];
        }
        const float inv = 1.f / (float)RN;
        feat[0] = n0 * inv; feat[1] = n1 * inv; feat[2] = n2 * inv;
        feat[3] = n3 * inv; feat[4] = n4 * inv;
        feat[5] = ne;
        feat[6] = ne / ((float)RN * (float)(RN - 1) * 0.5f);
        feat[7] = avg_deg;
        feat[8] = (poss > 0.f) ? (tri6 / poss) : 0.f;
        feat[9] = effsum / ((float)RN * (float)(RN - 1));
        feat[10] = (m2 > 0.f) ? (modsum / m2s) : 0.f;
    }
    __syncthreads();
    if (tid < 32) {
        float acc = b1[tid];
        for (int k = 0; k < 11; ++k) acc += W1[tid * 11 + k] * feat[k];
        h[tid] = fmaxf(acc, 0.f);
    }
    __syncthreads();
    if (tid < 64) {
        float acc = b2[tid];
        for (int k = 0; k < 32; ++k) acc += W2[tid * 32 + k] * h[k];
        outF[(size_t)band * 64 + tid] = acc;
    }
}

// ---------------------------------------------------------------------------
extern "C" void kernel_launch(void* const* d_in, const int* in_sizes, int n_in,
                              void* d_out, int out_size, void* d_ws, size_t ws_size,
                              hipStream_t stream) {
    const float* wc        = (const float*)d_in[0];
    // d_in[1] = frequency_bands (unused by reference math)
    const int*   community = (const int*)d_in[2];
    const float* W1        = (const float*)d_in[3];
    const float* b1        = (const float*)d_in[4];
    const float* W2        = (const float*)d_in[5];
    const float* b2        = (const float*)d_in[6];

    float* outA = (float*)d_out;                        // (32,5,96,96)
    float* outF = outA + (size_t)NBANDS * RN * RN;      // (32,5,64)

    float* G     = (float*)d_ws;                        // 160*9216 floats
    float* stats = G + (size_t)NBANDS * RN * RN;        // 15360*8 floats

    stats_kernel<<<NROWS, 256, 0, stream>>>(wc, stats);
    fft_entropy_kernel<<<NROWS, 256, 0, stream>>>(wc, stats);
    gram_kernel<<<NBANDS, 192, 0, stream>>>(wc, G);
    finalize_kernel<<<NBANDS, 256, 0, stream>>>(G, stats, community, W1, b1, W2, b2, outA, outF);
}